// query_attribute_value_20383914786894
// MI455X (gfx1250) — compile-verified
//
#include <hip/hip_runtime.h>

// ---------------------------------------------------------------------------
// Math collapse: softmax over a constant axis is uniform -> out[n,a,:] =
//   (mean_j attribute[j]) @ Wv^T + bv, then @ Wo^T + bo, replicated B*A times.
// Traffic: 256MB stores dominate (~11us @ 23.3TB/s). Everything else is tiny.
// ---------------------------------------------------------------------------

#define DIM   512
#define NB    1024          // attribute rows
#define NA    128           // prompt rows
#define OUT_FLOATS (NB * NA * DIM)   // 67,108,864 floats = 256 MiB

typedef __attribute__((ext_vector_type(2))) float v2f;
typedef __attribute__((ext_vector_type(4))) float v4f;
typedef __attribute__((ext_vector_type(8))) float v8f;

// ---- Kernel 1a: per-block partial column sums of attribute [1024 x 512] ----
// 16 blocks x 64 rows each; thread e accumulates column e (fully coalesced:
// at fixed j the 512 threads read one contiguous 2KB row).
__global__ __launch_bounds__(512) void colsum_partial(const float* __restrict__ A,
                                                      float* __restrict__ P) {
    const int e = threadIdx.x;               // column 0..511
    const int b = blockIdx.x;                // 0..15
    const float* row = A + (size_t)b * 64 * DIM;
    float s = 0.0f;
#pragma unroll 8
    for (int j = 0; j < 64; ++j)
        s += row[(size_t)j * DIM + e];
    P[b * DIM + e] = s;
}

// ---- Kernel 1b: deterministic fold of partials -> column mean --------------
__global__ __launch_bounds__(512) void fold_mean(const float* __restrict__ P,
                                                 float* __restrict__ abar) {
    const int e = threadIdx.x;
    float s = 0.0f;
#pragma unroll
    for (int b = 0; b < 16; ++b)
        s += P[b * DIM + e];
    abar[e] = s * (1.0f / (float)NB);
}

// ---- Kernel 2/3: y = W @ x + bias via V_WMMA_F32_16X16X4_F32 ---------------
// One wave per 16 output rows (32 blocks x 1 wave). Per step t:
//   A[m][k] = W[16*blk + m][4t + k]   (16x4 f32 fragment, 2 VGPRs/lane)
//   B[k][n] = x[4t + k] for all n     (columns identical -> D columns identical)
// Accumulate K=512 in 128 chained WMMAs (f32 accumulate, RNE).
__global__ __launch_bounds__(32) void matvec512_wmma(const float* __restrict__ W,
                                                     const float* __restrict__ x,
                                                     const float* __restrict__ bias,
                                                     float* __restrict__ y) {
    const int lane = threadIdx.x;        // 0..31, full wave -> EXEC all ones
    const int half = lane >> 4;          // K sub-block: 0 -> K={0,1}, 1 -> K={2,3}
    const int m    = lane & 15;          // A-matrix row within tile
    const int row  = blockIdx.x * 16 + m;

    const float* wrow = W + (size_t)row * DIM + 2 * half;  // 8B aligned
    const float* xh   = x + 2 * half;                      // 8B aligned

    v8f c = {0.f, 0.f, 0.f, 0.f, 0.f, 0.f, 0.f, 0.f};
#pragma unroll 8
    for (int t = 0; t < DIM / 4; ++t) {
        v2f a = *(const v2f*)(wrow + 4 * t);
        v2f b = *(const v2f*)(xh + 4 * t);
        // (neg_a, A, neg_b, B, c_mod, C, reuse_a, reuse_b)
        c = __builtin_amdgcn_wmma_f32_16x16x4_f32(false, a, false, b,
                                                  (short)0, c, false, false);
    }
    // D layout: lane 0 holds (M=0..7, N=0) in c[0..7]; lane 16 holds M=8..15.
    if (m == 0) {
        const int base = blockIdx.x * 16 + half * 8;
#pragma unroll
        for (int r = 0; r < 8; ++r)
            y[base + r] = c[r] + bias[base + r];
    }
}

// ---- Kernel 4: broadcast y (512 f) to all 131072 rows, NT b128 stores ------
// Grid stride (blocks*256 float4s) is a multiple of 128 float4s (= one row),
// so each thread's cached 16B of y is reused for every store it issues.
__global__ __launch_bounds__(256) void bcast_rows(const float* __restrict__ y,
                                                  float* __restrict__ out) {
    const int tid  = blockIdx.x * 256 + threadIdx.x;
    const int span = gridDim.x * 256;            // multiple of 128 by launch config
    const v4f val  = *(const v4f*)(y + 4 * (tid & 127));
    v4f* __restrict__ o4 = (v4f*)out;
    const int total4 = OUT_FLOATS / 4;           // 16,777,216
    for (int i = tid; i < total4; i += span)
        __builtin_nontemporal_store(val, o4 + i);   // 256MB > 192MB L2: stream it
}

extern "C" void kernel_launch(void* const* d_in, const int* in_sizes, int n_in,
                              void* d_out, int out_size, void* d_ws, size_t ws_size,
                              hipStream_t stream) {
    // setup_inputs() order: attribute, prompt, Wq, bq, Wk, bk, Wv, bv, Wo, bo
    const float* attribute = (const float*)d_in[0];
    const float* Wv        = (const float*)d_in[6];
    const float* bv        = (const float*)d_in[7];
    const float* Wo        = (const float*)d_in[8];
    const float* bo        = (const float*)d_in[9];
    float* out = (float*)d_out;

    float* ws   = (float*)d_ws;
    float* abar = ws;            // 512 floats: column mean of attribute
    float* vbar = ws + 512;      // 512 floats: mean of v rows
    float* yvec = ws + 1024;     // 512 floats: final replicated row
    float* P    = ws + 2048;     // 16*512 floats: partial column sums

    colsum_partial<<<16, 512, 0, stream>>>(attribute, P);
    fold_mean<<<1, 512, 0, stream>>>(P, abar);
    matvec512_wmma<<<32, 32, 0, stream>>>(Wv, abar, bv, vbar);
    matvec512_wmma<<<32, 32, 0, stream>>>(Wo, vbar, bo, yvec);
    bcast_rows<<<2048, 256, 0, stream>>>(yvec, out);
}